// NeuralCollapseRegularizer_30562987278503
// MI455X (gfx1250) — compile-verified
//
#include <hip/hip_runtime.h>
#include <hip/hip_bf16.h>

#define NROWS 131072
#define DDIM  512
#define NCLS  1000
#define ORDER_PAD 8

typedef __attribute__((ext_vector_type(2))) float v2f;
typedef __attribute__((ext_vector_type(8))) float v8f;

// ---------------------------------------------------------------- K0: zero counts + order pad
__global__ void nc_zero(int* __restrict__ counts, int* __restrict__ order) {
    int t = threadIdx.x;
    if (t < NCLS) counts[t] = 0;
    if (t < ORDER_PAD) order[NROWS + t] = 0;   // lets K4 prefetch unconditionally
}

// ---------------------------------------------------------------- K1: histogram
__global__ void nc_hist(const int* __restrict__ labels, int* __restrict__ counts) {
    int i = blockIdx.x * blockDim.x + threadIdx.x;
    if (i < NROWS) atomicAdd(&counts[labels[i]], 1);
}

// ---------------------------------------------------------------- K2: prefix scan (1 block)
__global__ void __launch_bounds__(1024) nc_scan(const int* __restrict__ counts,
                                                int* __restrict__ offsets,
                                                int* __restrict__ cursor) {
    __shared__ int s[1024];
    int t = threadIdx.x;
    int v = (t < NCLS) ? counts[t] : 0;
    s[t] = v;
    __syncthreads();
    for (int off = 1; off < 1024; off <<= 1) {
        int u = (t >= off) ? s[t - off] : 0;
        __syncthreads();
        s[t] += u;
        __syncthreads();
    }
    if (t < NCLS) {
        int e = s[t] - v;      // exclusive prefix
        offsets[t] = e;
        cursor[t]  = e;
    }
}

// ---------------------------------------------------------------- K3: scatter indices
__global__ void nc_scatter(const int* __restrict__ labels, int* __restrict__ cursor,
                           int* __restrict__ order) {
    int i = blockIdx.x * blockDim.x + threadIdx.x;
    if (i < NROWS) {
        int p = atomicAdd(&cursor[labels[i]], 1);
        order[p] = i;
    }
}

// ---------------------------------------------------------------- K4: per-class reduce (WMMA)
// One workgroup (8 waves) per class. Wave w owns dim-groups {4w..4w+3} (16 dims each,
// 8*4*16 = 512 = DDIM). Each chunk of 4 gathered rows is the 4x16 B matrix of
// V_WMMA_F32_16X16X4_F32 with A = ones(16x4): every accumulator row is the f32-exact
// column sum, so any row->laneslot placement inside B is valid.
// Main loop is guard-free and branchless (order[] is padded so the next-chunk prefetch
// never reads out of bounds); a single guarded tail chunk handles cnt%4.
__global__ void __launch_bounds__(256) nc_class_reduce(
        const float* __restrict__ X, const int* __restrict__ order,
        const int* __restrict__ counts, const int* __restrict__ offsets,
        float* __restrict__ sums, float* __restrict__ partial_sq) {
    const int c     = blockIdx.x;
    const int cnt   = counts[c];
    const int start = offsets[c];
    const int wave  = threadIdx.x >> 5;
    const int lane  = threadIdx.x & 31;
    const int half  = lane >> 4;
    const int n     = lane & 15;

    v2f ones = {1.0f, 1.0f};
    v8f acc[4] = {};
    float sq = 0.0f;

    const int full = cnt & ~3;           // block-uniform

#pragma unroll 2
    for (int r = 0; r < full; r += 4) {
        const int i0 = order[start + r];
        const int i1 = order[start + r + 1];
        const int i2 = order[start + r + 2];
        const int i3 = order[start + r + 3];

        // lanes 0-15 cover rows (i0,i1); lanes 16-31 cover rows (i2,i3)
        const float* rowA = X + (size_t)(half ? i2 : i0) * DDIM;
        const float* rowB = X + (size_t)(half ? i3 : i1) * DDIM;

        // unconditional: order[] is padded, worst case we prefetch row 0
        const int nx = order[start + r + 4];
        __builtin_prefetch(X + (size_t)nx * DDIM + wave * 64, 0, 1);

#pragma unroll
        for (int j = 0; j < 4; ++j) {
            const int base = (wave * 4 + j) * 16;
            float bx = rowA[base + n];
            float by = rowB[base + n];
            sq += bx * bx + by * by;      // VALU co-executes with XDL wmma
            v2f b = {bx, by};
            acc[j] = __builtin_amdgcn_wmma_f32_16x16x4_f32(
                false, ones, false, b, (short)0, acc[j], false, false);
        }
    }

    if (full < cnt) {                    // tail chunk, executed at most once
        const int  i0 = order[start + full];
        const bool v1 = (full + 1 < cnt), v2 = (full + 2 < cnt), v3 = (full + 3 < cnt);
        const int  i1 = v1 ? order[start + full + 1] : i0;
        const int  i2 = v2 ? order[start + full + 2] : i0;
        const int  i3 = v3 ? order[start + full + 3] : i0;

        const float* rowA = X + (size_t)(half ? i2 : i0) * DDIM;
        const float* rowB = X + (size_t)(half ? i3 : i1) * DDIM;
        const bool   vA   = half ? v2 : true;
        const bool   vB   = half ? v3 : v1;

#pragma unroll
        for (int j = 0; j < 4; ++j) {
            const int base = (wave * 4 + j) * 16;
            float bx = vA ? rowA[base + n] : 0.0f;
            float by = vB ? rowB[base + n] : 0.0f;
            sq += bx * bx + by * by;
            v2f b = {bx, by};
            acc[j] = __builtin_amdgcn_wmma_f32_16x16x4_f32(
                false, ones, false, b, (short)0, acc[j], false, false);
        }
    }

    // D-matrix VGPR0, lanes 0-15: M=0, N=lane -> column sum for dim base+lane
    if (lane < 16) {
#pragma unroll
        for (int j = 0; j < 4; ++j)
            sums[(size_t)c * DDIM + (wave * 4 + j) * 16 + lane] = acc[j][0];
    }

    // block-reduce sum of squares for this class
    __shared__ float red[256];
    red[threadIdx.x] = sq;
    __syncthreads();
    for (int s2 = 128; s2 > 0; s2 >>= 1) {
        if (threadIdx.x < s2) red[threadIdx.x] += red[threadIdx.x + s2];
        __syncthreads();
    }
    if (threadIdx.x == 0) partial_sq[c] = red[0];
}

// ---------------------------------------------------------------- K5: finalize (1 block)
__global__ void __launch_bounds__(1024) nc_finalize(
        const float* __restrict__ sums, const int* __restrict__ counts,
        const float* __restrict__ partial_sq, float* __restrict__ out) {
    __shared__ float  gmean[DDIM];
    __shared__ double red[1024];
    const int t = threadIdx.x;

    // global mean: 1024 threads, each sums one column over one 500-class half
    {
        const int d = t & (DDIM - 1);
        const int h = t >> 9;            // 0 or 1
        double g = 0.0;
        for (int c = h * 500; c < h * 500 + 500; ++c)
            g += (double)sums[c * DDIM + d];
        red[t] = g;
        __syncthreads();
        if (t < DDIM)
            gmean[t] = (float)((red[t] + red[t + DDIM]) / (double)NROWS);
        __syncthreads();
    }

    double l_sb = 0.0, l_w = 0.0, l_sq = 0.0;
    if (t < NCLS) {
        float cntf = (float)counts[t];
        l_sq = (double)partial_sq[t];
        for (int d = 0; d < DDIM; ++d) {
            float s    = sums[t * DDIM + d];
            float mean = s / cntf;
            float dm   = mean - gmean[d];
            l_sb += (double)dm * (double)dm;
            l_w  += (double)mean * (double)s;   // n_c * ||mu_c||^2 contribution
        }
    }

    red[t] = l_sb; __syncthreads();
    for (int s2 = 512; s2 > 0; s2 >>= 1) { if (t < s2) red[t] += red[t + s2]; __syncthreads(); }
    double tr_sb = red[0]; __syncthreads();

    red[t] = l_w; __syncthreads();
    for (int s2 = 512; s2 > 0; s2 >>= 1) { if (t < s2) red[t] += red[t + s2]; __syncthreads(); }
    double w_term = red[0]; __syncthreads();

    red[t] = l_sq; __syncthreads();
    for (int s2 = 512; s2 > 0; s2 >>= 1) { if (t < s2) red[t] += red[t + s2]; __syncthreads(); }

    if (t == 0) {
        double tr_sw = red[0] - w_term;   // sum||x||^2 - sum_c n_c ||mu_c||^2
        out[0] = (float)(tr_sw / tr_sb);
    }
}

// ---------------------------------------------------------------- launch
extern "C" void kernel_launch(void* const* d_in, const int* in_sizes, int n_in,
                              void* d_out, int out_size, void* d_ws, size_t ws_size,
                              hipStream_t stream) {
    const float* X      = (const float*)d_in[0];
    const int*   labels = (const int*)d_in[1];
    float*       out    = (float*)d_out;

    char* ws = (char*)d_ws;
    int*   counts     = (int*)(ws);
    int*   offsets    = (int*)(ws + 4096);
    int*   cursor     = (int*)(ws + 8192);
    float* partial_sq = (float*)(ws + 12288);
    int*   order      = (int*)(ws + 16384);
    float* sums       = (float*)(ws + 16384 + (size_t)(NROWS + ORDER_PAD) * 4);

    nc_zero<<<1, 1024, 0, stream>>>(counts, order);
    nc_hist<<<NROWS / 256, 256, 0, stream>>>(labels, counts);
    nc_scan<<<1, 1024, 0, stream>>>(counts, offsets, cursor);
    nc_scatter<<<NROWS / 256, 256, 0, stream>>>(labels, cursor, order);
    nc_class_reduce<<<NCLS, 256, 0, stream>>>(X, order, counts, offsets, sums, partial_sq);
    nc_finalize<<<1, 1024, 0, stream>>>(sums, counts, partial_sq, out);
}